// EmbeddingLoss_70145405878948
// MI455X (gfx1250) — compile-verified
//
#include <hip/hip_runtime.h>
#include <hip/hip_bf16.h>

typedef __attribute__((ext_vector_type(2))) float v2f;
typedef __attribute__((ext_vector_type(8))) float v8f;

static constexpr int    Nn   = 2;
static constexpr int    Cc   = 8;
static constexpr int    Ii   = 8;
static constexpr size_t Pp   = 8ull * 256ull * 448ull;   // 917504 pixels per batch
static constexpr int    BINS = 65536;
static constexpr int    NSEG = 16;                        // N * I segments

// ---- workspace layout (bytes) ----
static constexpr size_t OFF_H64 = 0;                               // u64 hist: (count<<32)|label_sum
static constexpr size_t SZ_H64  = (size_t)NSEG * BINS * 8;
static constexpr size_t OFF_ST  = OFF_H64 + SZ_H64;                // raw stats [seg][16]
static constexpr size_t SZ_ST   = (size_t)NSEG * 16 * 4;
static constexpr size_t OFF_CF  = OFF_ST + SZ_ST;                  // WMMA B-matrix coefs [n][3][4][16]
static constexpr size_t SZ_CF   = (size_t)Nn * 3 * 4 * 16 * 4;
static constexpr size_t OFF_AC  = OFF_CF + SZ_CF;                  // accumulators (96 floats)
static constexpr size_t SZ_AC   = 96 * 4;
static constexpr size_t ZERO_WORDS = (OFF_AC + SZ_AC) / 4;
// accum slots: [0..15] lov  [16..31] instterm  [32..47] cnt_s  [48..63] valid
//              [64..79] inst_sm  [80..81] bgterm  [82..83] bgcnt

__device__ __forceinline__ float wave_sum(float v) {
#pragma unroll
  for (int o = 16; o > 0; o >>= 1) v += __shfl_xor(v, o, 32);
  return v;
}

// ---------------- 0) zero workspace ----------------
__global__ void el_zero_kernel(unsigned* ws, size_t nwords) {
  for (size_t i = (size_t)blockIdx.x * blockDim.x + threadIdx.x; i < nwords;
       i += (size_t)gridDim.x * blockDim.x)
    ws[i] = 0u;
}

// ---------------- 1) per-instance raw statistics ----------------
// slots: 0=cnt, 1..4=sum(m*emb_e), 5..7=sum(m*exp(bw)*10), 8..10=sum(m*bw), 11..13=sum(m*bw^2)
__global__ void el_stats_kernel(const float* __restrict__ emb,
                                const int* __restrict__ masks,
                                float* __restrict__ stats) {
  const int n = blockIdx.z, i = blockIdx.y;
  const int* mrow = masks + ((size_t)n * Ii + i) * Pp;
  const float* eb = emb + (size_t)n * Cc * Pp;
  float acc[14];
#pragma unroll
  for (int k = 0; k < 14; k++) acc[k] = 0.f;
  for (size_t p = (size_t)blockIdx.x * blockDim.x + threadIdx.x; p < Pp;
       p += (size_t)gridDim.x * blockDim.x) {
    if (mrow[p] != 0) {
      acc[0] += 1.f;
#pragma unroll
      for (int e = 0; e < 4; e++) acc[1 + e] += eb[(size_t)e * Pp + p];
#pragma unroll
      for (int b = 0; b < 3; b++) {
        float w = eb[(size_t)(4 + b) * Pp + p];
        acc[5 + b]  += __expf(w) * 10.f;
        acc[8 + b]  += w;
        acc[11 + b] += w * w;
      }
    }
  }
  const int seg = n * Ii + i;
#pragma unroll
  for (int k = 0; k < 14; k++) {
    float s = wave_sum(acc[k]);
    if ((threadIdx.x & 31) == 0) atomicAdd(&stats[(size_t)seg * 16 + k], s);
  }
}

// ---------------- 2) finalize: build WMMA B-matrix coefficients ----------------
__global__ void el_finalize_kernel(const float* __restrict__ stats,
                                   float* __restrict__ coef,
                                   float* __restrict__ accum) {
  const int seg = threadIdx.x;
  if (seg >= NSEG) return;
  const int n = seg >> 3, i = seg & 7;
  const float* s = stats + (size_t)seg * 16;
  const float cnt   = s[0];
  const float cnt_s = fmaxf(cnt, 1.f);
  const float valid = (cnt > 0.f) ? 1.f : 0.f;
  float mu[4], mbf[4];
#pragma unroll
  for (int e = 0; e < 4; e++) mu[e] = s[1 + e] / cnt_s;
#pragma unroll
  for (int b = 0; b < 3; b++) mbf[b] = s[5 + b] / cnt_s;
  mbf[3] = 1.0f;  // FREE_BW
  float t3 = 0.f;
#pragma unroll
  for (int e = 0; e < 4; e++) t3 += mu[e] * mu[e] * mbf[e];
  // coef[((n*3+mat)*4+k)*16 + col]; cols >= 8 stay zero (padding)
#pragma unroll
  for (int k = 0; k < 4; k++) {
    coef[((size_t)(n * 3 + 0) * 4 + k) * 16 + i] = mbf[k];
    coef[((size_t)(n * 3 + 1) * 4 + k) * 16 + i] = -2.f * mu[k] * mbf[k];
    coef[((size_t)(n * 3 + 2) * 4 + k) * 16 + i] = (k == 0) ? t3 : 0.f;
  }
  float vs = 0.f;
#pragma unroll
  for (int b = 0; b < 3; b++) {
    float mbr = s[8 + b] / cnt_s;
    vs += s[11 + b] - cnt_s * mbr * mbr;
  }
  accum[32 + seg] = cnt_s;
  accum[48 + seg] = valid;
  accum[64 + seg] = vs / (cnt_s * 3.0f);   // inst_sm
}

// ---------------- 3) WMMA per-pixel pass ----------------
// dist[p, i] via V_WMMA_F32_16X16X4_F32 chain; then exp/logit/error -> histogram,
// plus seed instance term and background term (branchless bookkeeping).
__global__ void el_pixel_kernel(const float* __restrict__ emb,
                                const int* __restrict__ masks,
                                const unsigned char* __restrict__ ign,
                                const float* __restrict__ coef,
                                unsigned long long* __restrict__ h64,
                                float* __restrict__ accum) {
  const int n = blockIdx.y;
  const float* eb = emb + (size_t)n * Cc * Pp;
  const int lane  = threadIdx.x & 31;
  const int wave  = threadIdx.x >> 5;
  const int col   = lane & 15;               // B/D column = instance
  const int kbase = (lane < 16) ? 0 : 2;     // K pair held by this lane

  // B matrices (wave-constant, per batch)
  const float* cb = coef + (size_t)n * 3 * 4 * 16;
  v2f B1 = { cb[(0 * 4 + kbase) * 16 + col], cb[(0 * 4 + kbase + 1) * 16 + col] };
  v2f B2 = { cb[(1 * 4 + kbase) * 16 + col], cb[(1 * 4 + kbase + 1) * 16 + col] };
  v2f B3 = { cb[(2 * 4 + kbase) * 16 + col], cb[(2 * 4 + kbase + 1) * 16 + col] };
  const v2f A3 = { (lane < 16) ? 1.f : 0.f, 0.f };   // selects K=0 row of B3 (t3 broadcast)

  float i_term = 0.f, bgterm = 0.f, bgcnt = 0.f;
  const bool  act  = (col < 8);
  const float wsel = (lane == 0 || lane == 16) ? 1.f : 0.f;  // bg bookkeeping weight
  const int   off  = (lane < 16) ? 0 : 8;    // D rows: pixels p0+off+r
  // always-valid mask row (padding lanes read a clamped row and ignore it)
  const int* mrow = masks + ((size_t)n * Ii + (col & 7)) * Pp;
  const unsigned char* irow = ign + (size_t)n * Pp;
  const float* seedp = eb + (size_t)7 * Pp;

  const int NT = (int)(Pp / 16);
  const int nw = gridDim.x * (blockDim.x >> 5);
  const int pfstride = 16 * nw;              // next grid-stride iteration distance
  for (int t = blockIdx.x * (blockDim.x >> 5) + wave; t < NT; t += nw) {
    const int p0 = t * 16;
    // A: features of pixel p0+col for K pair (kbase, kbase+1)
    const int pixA = p0 + col;
    const float a0 = eb[(size_t)kbase * Pp + pixA];
    const float a1 = eb[(size_t)(kbase + 1) * Pp + pixA];
    __builtin_prefetch(eb + (size_t)kbase * Pp + pixA + pfstride, 0, 1);
    v2f A1 = { a0 * a0, a1 * a1 };
    v2f A2 = { a0, a1 };
    v8f d = {};
    d = __builtin_amdgcn_wmma_f32_16x16x4_f32(false, A3, false, B3, (short)0, d, false, false);
    d = __builtin_amdgcn_wmma_f32_16x16x4_f32(false, A2, false, B2, (short)0, d, false, false);
    d = __builtin_amdgcn_wmma_f32_16x16x4_f32(false, A1, false, B1, (short)0, d, false, false);

    const int pbase = p0 + off;
    const int4 m0 = *(const int4*)(mrow + pbase);
    const int4 m1 = *(const int4*)(mrow + pbase + 4);
    __builtin_prefetch(mrow + pbase + pfstride, 0, 1);
    const int mv[8] = { m0.x, m0.y, m0.z, m0.w, m1.x, m1.y, m1.z, m1.w };
    const uchar4 ig0 = *(const uchar4*)(irow + pbase);
    const uchar4 ig1 = *(const uchar4*)(irow + pbase + 4);
    const unsigned char igb[8] = { ig0.x, ig0.y, ig0.z, ig0.w,
                                   ig1.x, ig1.y, ig1.z, ig1.w };
    float sd[8];
#pragma unroll
    for (int r = 0; r < 8; r++) sd[r] = seedp[pbase + r];  // seed channel

#pragma unroll
    for (int r = 0; r < 8; r++) {
      const float pr = __expf(-0.5f * d[r]);
      const unsigned long long bal = __ballot(act && (mv[r] != 0));
      if (act) {
        const float mf    = (float)mv[r];
        const float err   = 1.f - (2.f * pr - 1.f) * (2.f * mf - 1.f);  // in [0,2)
        int bin = (int)(err * 32768.f);
        bin = (bin < 0) ? 0 : ((bin > BINS - 1) ? BINS - 1 : bin);
        const int seg = n * Ii + col;
        atomicAdd(&h64[(size_t)seg * BINS + bin],
                  (1ull << 32) | (unsigned long long)(unsigned)mv[r]);
        const float ds = sd[r] - pr;
        i_term += mf * ds * ds;
      }
      // branchless background bookkeeping: only lanes 0 and 16 carry weight
      const unsigned bits = (lane < 16) ? (unsigned)(bal & 0xFFull)
                                        : (unsigned)((bal >> 16) & 0xFFull);
      const float isbg = (bits == 0u) ? wsel : 0.f;
      bgcnt += isbg;
      const float keep = igb[r] ? 0.f : isbg;
      bgterm += keep * sd[r] * sd[r];
    }
  }
  i_term += __shfl_xor(i_term, 16, 32);             // combine the two half-wave pixel groups
  if (lane < 8) atomicAdd(&accum[16 + n * Ii + lane], i_term);
  if (lane == 0 || lane == 16) {
    atomicAdd(&accum[80 + n], bgterm);
    atomicAdd(&accum[82 + n], bgcnt);
  }
}

// ---------------- 4) Lovász hinge via descending bin walk ----------------
__global__ void el_lovasz_kernel(const unsigned long long* __restrict__ h64,
                                 const float* __restrict__ stats,
                                 float* __restrict__ accum) {
  __shared__ unsigned sc[256], sg[256];
  __shared__ float sf[8];
  const int seg = blockIdx.x;
  const unsigned long long* h = h64 + (size_t)seg * BINS;
  const float G = stats[(size_t)seg * 16 + 0];
  const int tid = threadIdx.x;
  const int CH  = BINS / 256;
  const int top = BINS - 1 - tid * CH;  // descending chunk [top .. top-CH+1]

  unsigned lc = 0, lg = 0;
  for (int k = 0; k < CH; k++) {
    unsigned long long v = h[top - k];
    lc += (unsigned)(v >> 32);
    lg += (unsigned)v;
  }
  sc[tid] = lc; sg[tid] = lg;
  __syncthreads();
  for (int ofs = 1; ofs < 256; ofs <<= 1) {
    unsigned ac = (tid >= ofs) ? sc[tid - ofs] : 0u;
    unsigned ag = (tid >= ofs) ? sg[tid - ofs] : 0u;
    __syncthreads();
    sc[tid] += ac; sg[tid] += ag;
    __syncthreads();
  }
  unsigned j = sc[tid] - lc;   // exclusive prefix: elements strictly above my chunk
  unsigned c = sg[tid] - lg;

  float acc = 0.f;
  for (int k = 0; k < CH; k++) {
    const int b = top - k;
    unsigned long long v = h[b];
    const unsigned cbn = (unsigned)(v >> 32);
    if (cbn) {
      const unsigned gbn = (unsigned)v;
      const float F0 = (j == 0) ? 0.f
                     : 1.f - (G - (float)c) / (G + (float)j - (float)c);
      j += cbn; c += gbn;
      const float F1 = 1.f - (G - (float)c) / (G + (float)j - (float)c);
      const float e = ((float)b + 0.5f) * (1.0f / 32768.0f);  // bin midpoint
      acc += e * (F1 - F0);   // grouped (equal-error) jaccard telescoping
    }
  }
  float ws = wave_sum(acc);
  if ((tid & 31) == 0) sf[tid >> 5] = ws;
  __syncthreads();
  if (tid == 0) {
    float tot = 0.f;
    for (int w = 0; w < 8; w++) tot += sf[w];
    accum[seg] = tot;
  }
}

// ---------------- 5) scalar combine ----------------
__global__ void el_final_kernel(const float* __restrict__ accum,
                                float* __restrict__ out) {
  float totvalid = 0.f;
  for (int s = 0; s < NSEG; s++) totvalid += accum[48 + s];
  const float total = fmaxf(totvalid, 1.f);
  float lov = 0.f;
  for (int s = 0; s < NSEG; s++) lov += accum[48 + s] * accum[s];
  const float lovasz_loss = lov / total;
  float bsm = 0.f;
  for (int n = 0; n < Nn; n++) {
    float ni = 0.f, sum = 0.f;
    for (int i = 0; i < Ii; i++) {
      const int s = n * Ii + i;
      ni  += accum[48 + s];
      sum += accum[48 + s] * accum[64 + s];
    }
    bsm += sum / fmaxf(ni, 1.f);
  }
  bsm /= (float)Nn;
  float bg = 0.f;
  for (int n = 0; n < Nn; n++) bg += accum[80 + n] / fmaxf(accum[82 + n], 1.f);
  float it = 0.f;
  for (int s = 0; s < NSEG; s++) it += accum[48 + s] * (accum[16 + s] / accum[32 + s]);
  const float seed_loss = (bg + it) / (total + 1.f);
  out[0] = lovasz_loss * 1.0f + bsm * 10.0f + seed_loss * 1.0f;
}

extern "C" void kernel_launch(void* const* d_in, const int* in_sizes, int n_in,
                              void* d_out, int out_size, void* d_ws, size_t ws_size,
                              hipStream_t stream) {
  const float*         emb   = (const float*)d_in[0];
  const int*           masks = (const int*)d_in[1];
  const unsigned char* ign   = (const unsigned char*)d_in[2];  // jax bool -> 1 byte/elem
  float* out = (float*)d_out;

  char* ws = (char*)d_ws;
  unsigned long long* h64 = (unsigned long long*)(ws + OFF_H64);
  float* stats = (float*)(ws + OFF_ST);
  float* coef  = (float*)(ws + OFF_CF);
  float* accum = (float*)(ws + OFF_AC);

  el_zero_kernel<<<2048, 256, 0, stream>>>((unsigned*)ws, ZERO_WORDS);
  el_stats_kernel<<<dim3(256, Ii, Nn), 256, 0, stream>>>(emb, masks, stats);
  el_finalize_kernel<<<1, 32, 0, stream>>>(stats, coef, accum);
  el_pixel_kernel<<<dim3(896, Nn), 256, 0, stream>>>(emb, masks, ign, coef, h64, accum);
  el_lovasz_kernel<<<NSEG, 256, 0, stream>>>(h64, stats, accum);
  el_final_kernel<<<1, 1, 0, stream>>>(accum, out);
}